// InitReduceConv_89163521065167
// MI455X (gfx1250) — compile-verified
//
#include <hip/hip_runtime.h>

// ---------------------------------------------------------------------------
// Zero-fill the output (harness poisons d_out with 0xAA before timing).
// ---------------------------------------------------------------------------
__global__ __launch_bounds__(256) void zero_f32_kernel(float* __restrict__ out, int n) {
    int i      = blockIdx.x * blockDim.x + threadIdx.x;
    int stride = gridDim.x * blockDim.x;
    int n4     = n >> 2;
    float4* o4 = reinterpret_cast<float4*>(out);
    float4 z;  z.x = 0.f; z.y = 0.f; z.z = 0.f; z.w = 0.f;
    for (int k = i; k < n4; k += stride) o4[k] = z;               // global_store_b128
    for (int k = (n4 << 2) + i; k < n; k += stride) out[k] = 0.f; // tail
}

// ---------------------------------------------------------------------------
// Gather + scatter-add, double-buffered async index staging.
//   x      : [n_src, 64] fp32          (25.6 MB -> L2 resident)
//   srcIdx : [n_edges] int32 (gather rows)
//   dstIdx : [n_edges] int32 (scatter rows)
//   out    : [n_out, 64] fp32 (pre-zeroed, L2 resident)
//
// One wave == one 32-edge batch (grid-stride). Batch k+1's indices are DMA'd
// into LDS (global_load_async_to_lds_b32, ASYNCcnt) while batch k computes;
// s_wait_asynccnt 2 retires the current batch's DMA (in-order completion).
// Per edge: one coalesced 256B row gather (global_load_b64 x32 lanes) and
// 64 no-return global_atomic_add_f32 resolved in L2.
// ---------------------------------------------------------------------------
#define TPB 256
#define WAVES_PER_BLOCK (TPB / 32)

__global__ __launch_bounds__(TPB) void gather_scatter_add_kernel(
    const float* __restrict__ x,
    const int*   __restrict__ srcIdx,
    const int*   __restrict__ dstIdx,
    float*       __restrict__ out,
    int n_edges)
{
    // [buffer][wave][0=src,1=dst][lane]; 128B-aligned slices -> b128 LDS reads ok
    __shared__ __align__(16) int sIdx[2][WAVES_PER_BLOCK][2][32];

    const int lane      = threadIdx.x & 31;            // wave32
    const int waveInBlk = threadIdx.x >> 5;
    const int gwave     = blockIdx.x * WAVES_PER_BLOCK + waveInBlk;
    const int nwaves    = gridDim.x * WAVES_PER_BLOCK;
    const int stride    = nwaves * 32;

    // LDS byte offsets of this lane's staging slots (generic->LDS keeps the
    // segment-relative offset in the low 32 bits, ISA 10.2).
    unsigned offS[2], offD[2];
    offS[0] = (unsigned)(size_t)&sIdx[0][waveInBlk][0][lane];
    offD[0] = (unsigned)(size_t)&sIdx[0][waveInBlk][1][lane];
    offS[1] = (unsigned)(size_t)&sIdx[1][waveInBlk][0][lane];
    offD[1] = (unsigned)(size_t)&sIdx[1][waveInBlk][1][lane];

    // Async-DMA 32 src + 32 dst indices for batch at `b` into buffer `buf`.
    // Keeps "memory" clobber: acts as compiler fence so LDS reads of the
    // buffer being overwritten can never sink below the restage.
    auto stage = [&](int buf, int b) {
        int e = b + lane;
        if (e >= n_edges) e = n_edges - 1;   // clamp; tail lanes unused
        const int* gs = srcIdx + e;
        const int* gd = dstIdx + e;
        asm volatile("global_load_async_to_lds_b32 %0, %1, off"
                     :: "v"(offS[buf]), "v"(gs) : "memory");
        asm volatile("global_load_async_to_lds_b32 %0, %1, off"
                     :: "v"(offD[buf]), "v"(gd) : "memory");
    };

    // One edge: coalesced float2 gather + two no-return f32 atomics.
    // NOTE: no "memory" clobber -> compiler may hoist later gathers above
    // these atomics (data deps already order each gather->atomic pair).
    auto edgeOp = [&](int s, int d) {
        const float2 v = *reinterpret_cast<const float2*>(
            x + ((size_t)s << 6) + (lane << 1));
        float* p = out + ((size_t)d << 6) + (lane << 1);
        asm volatile("global_atomic_add_f32 %0, %1, off" :: "v"(p),     "v"(v.x));
        asm volatile("global_atomic_add_f32 %0, %1, off" :: "v"(p + 1), "v"(v.y));
    };

    int base = gwave * 32;
    if (base >= n_edges) return;

    int buf = 0;
    stage(0, base);                                   // prologue DMA

    for (; base < n_edges; base += stride) {
        // Stage next batch (or harmlessly restage current if at the tail) so
        // the wait immediate stays a constant 2.
        int nextBase = base + stride;
        stage(buf ^ 1, (nextBase < n_edges) ? nextBase : base);

        // Current batch's 2 DMAs are the oldest outstanding -> done at cnt<=2.
        asm volatile("s_wait_asynccnt 0x2" ::: "memory");

        const int* mySrc = &sIdx[buf][waveInBlk][0][0];
        const int* myDst = &sIdx[buf][waveInBlk][1][0];

        // Warm the 32 gather rows (256B each = 2 cachelines).
        {
            const int sPre = mySrc[lane];
            const float* rp = x + ((size_t)sPre << 6);
            __builtin_prefetch(rp, 0, 3);        // global_prefetch_b8
            __builtin_prefetch(rp + 32, 0, 3);
        }

        const int cnt = (n_edges - base < 32) ? (n_edges - base) : 32;

        if (cnt == 32) {
            // 4-edge chunks: one ds_load_b128 pair feeds 4 independent
            // gather+atomic streams (good MLP, 4x fewer DS ops).
            for (int jj = 0; jj < 32; jj += 4) {
                const int4 s4 = *reinterpret_cast<const int4*>(mySrc + jj);
                const int4 d4 = *reinterpret_cast<const int4*>(myDst + jj);
                edgeOp(s4.x, d4.x);
                edgeOp(s4.y, d4.y);
                edgeOp(s4.z, d4.z);
                edgeOp(s4.w, d4.w);
            }
        } else {
            for (int j = 0; j < cnt; ++j) edgeOp(mySrc[j], myDst[j]);
        }

        buf ^= 1;
    }
}

// ---------------------------------------------------------------------------
// Harness entry point.
// Inputs (setup_inputs order): boundary_x [N_SRC*64] f32,
//                              boundary_index [2*N_EDGES] int,
//                              out_size scalar.
// Output: [N_OUT * 64] f32, flat in d_out (out_size elements).
// ---------------------------------------------------------------------------
extern "C" void kernel_launch(void* const* d_in, const int* in_sizes, int n_in,
                              void* d_out, int out_size, void* d_ws, size_t ws_size,
                              hipStream_t stream) {
    const float* x   = (const float*)d_in[0];
    const int*   idx = (const int*)d_in[1];
    float*       out = (float*)d_out;

    const int n_edges = in_sizes[1] / 2;
    const int* srcIdx = idx;            // boundary_index[0]
    const int* dstIdx = idx + n_edges;  // boundary_index[1]

    // 1) zero the output
    int zBlocks = (out_size / 4 + TPB - 1) / TPB;
    if (zBlocks < 1) zBlocks = 1;
    if (zBlocks > 16384) zBlocks = 16384;
    zero_f32_kernel<<<zBlocks, TPB, 0, stream>>>(out, out_size);

    // 2) gather + atomic scatter-add (grid-stride over 32-edge wave batches)
    int blocks = 2048;                  // 16384 waves; ~4 batches each at 2M edges
    gather_scatter_add_kernel<<<blocks, TPB, 0, stream>>>(x, srcIdx, dstIdx, out, n_edges);
}